// LinearChainCRF_42356967473702
// MI455X (gfx1250) — compile-verified
//
#include <hip/hip_runtime.h>

#define NB   16384
#define NT   512
#define NTAG 4
#define LN2  0.69314718055994530942f

typedef __attribute__((ext_vector_type(2))) float v2f;
typedef __attribute__((ext_vector_type(8))) float v8f;

// One wave processes 16 sequences. Lane l and lane l+16 both handle sequence
// n = l&15 (duplicate addresses coalesce to the same cacheline -> no extra
// DRAM traffic). A rows 8..11 replicate rows 0..3, so after each WMMA every
// lane holds all 4 beta values of its own sequence in d[0..3] -- no shuffles.
//
// Probability-space forward recursion with per-step renormalization:
//   beta_t = rcp(sum) * ( (M^T beta_{t-1}) .* exp(emit_t) ),  logz2 += log2(sum)
// M = exp(transitions).  D = A x B via V_WMMA_F32_16X16X4_F32 with
// A[16x4] = M^T (rows {0-3,8-11}), B[4x16] = beta block of 16 sequences.

struct CrfState {
    float p0, p1, p2, p3;   // normalized beta (per-lane, this lane's sequence)
    float logz2;            // accumulated log2(scale)
    float gold;             // gold path score
    int   tprev;            // previous tag
};

// Branchless 4-way mux on tag bits (lowers to v_cndmask, no exec branches).
__device__ __forceinline__ float pick4(float4 e, int tc) {
    bool b0 = (tc & 1) != 0;
    bool b1 = (tc & 2) != 0;
    float lo = b0 ? e.y : e.x;
    float hi = b0 ? e.w : e.z;
    return b1 ? hi : lo;
}

__device__ __forceinline__ void crf_step(CrfState& st, const v2f a, int lane,
                                         float4 et, int tc, int mm,
                                         const float* __restrict__ s_trans)
{
    // B operand: B[k,n] = p_k of sequence n (VGPR0 = K0|K2, VGPR1 = K1|K3).
    v2f b;
    b.x = (lane < 16) ? st.p0 : st.p2;
    b.y = (lane < 16) ? st.p1 : st.p3;
    v8f c = {};
    v8f d = __builtin_amdgcn_wmma_f32_16x16x4_f32(
                false, a, false, b, (short)0, c, false, false);

    float v0 = d[0] * __expf(et.x);
    float v1 = d[1] * __expf(et.y);
    float v2 = d[2] * __expf(et.z);
    float v3 = d[3] * __expf(et.w);

    float ss = (v0 + v1) + (v2 + v3);
    float f  = __builtin_amdgcn_rcpf(ss);     // v_rcp_f32 (no IEEE div tail)
    float l2 = __builtin_amdgcn_logf(ss);     // v_log_f32 = log2(ss), raw

    // Branchless masked update (keeps wave convergent around the WMMA).
    bool um = (mm != 0);
    st.p0 = um ? v0 * f : st.p0;
    st.p1 = um ? v1 * f : st.p1;
    st.p2 = um ? v2 * f : st.p2;
    st.p3 = um ? v3 * f : st.p3;
    st.logz2 += um ? l2 : 0.0f;

    float mf = um ? 1.0f : 0.0f;
    float pe = pick4(et, tc);
    st.gold += (pe + s_trans[st.tprev * 4 + tc]) * mf;
    st.tprev = tc;
}

__global__ __launch_bounds__(256) void crf_fwd_kernel(
    const float* __restrict__ em,            // [B,T,4]
    const int*   __restrict__ tags,          // [B,T]
    const unsigned char* __restrict__ mask,  // [B,T] (bool)
    const float* __restrict__ trans,         // [4,4]
    float* __restrict__ per_seq)             // [B]  logZ - gold
{
    __shared__ float s_trans[16];
    __shared__ float s_expT[16];

    const int tid = threadIdx.x;
    if (tid < 16) {
        float tv = trans[tid];
        s_trans[tid] = tv;
        s_expT[tid]  = __expf(tv);
    }
    __syncthreads();

    const int lane = tid & 31;
    const int wave = tid >> 5;
    const int n    = lane & 15;
    const int seq  = blockIdx.x * 128 + wave * 16 + n;

    // A operand: A[m,k] = expT[k][m&3] for m in {0..3, 8..11}, else 0.
    // 32-bit A 16x4 layout: M = lane&15; VGPR0 = K0 (lanes 0-15) | K2
    // (lanes 16-31); VGPR1 = K1 | K3.
    const int  m    = lane & 15;
    const int  k0   = (lane < 16) ? 0 : 2;
    const bool mrow = ((m & 4) == 0);
    v2f a;
    a.x = mrow ? s_expT[(k0    ) * 4 + (m & 3)] : 0.0f;
    a.y = mrow ? s_expT[(k0 + 1) * 4 + (m & 3)] : 0.0f;

    const float4*        ep4 = (const float4*)(em + (size_t)seq * (NT * NTAG));
    const int*           tp  = tags + (size_t)seq * NT;
    const unsigned char* mp  = mask + (size_t)seq * NT;

    CrfState st;

    // ---- first batch of 4 timesteps (t=0 init + steps 1..3)
    {
        float4   ea = ep4[0], eb = ep4[1], ec = ep4[2], ed = ep4[3];
        int4     tg = *(const int4*)(tp);
        unsigned mw = *(const unsigned*)(mp);

        // t = 0: beta0 = normalize(exp(em0)); forward init ignores mask.
        float q0 = __expf(ea.x), q1 = __expf(ea.y), q2 = __expf(ea.z), q3 = __expf(ea.w);
        float s0 = (q0 + q1) + (q2 + q3);
        float f0 = __builtin_amdgcn_rcpf(s0);
        st.p0 = q0 * f0; st.p1 = q1 * f0; st.p2 = q2 * f0; st.p3 = q3 * f0;
        st.logz2 = __builtin_amdgcn_logf(s0);
        st.tprev = tg.x;
        float mf0 = (mw & 0xff) ? 1.0f : 0.0f;
        st.gold   = pick4(ea, tg.x) * mf0;

        crf_step(st, a, lane, eb, tg.y, (int)((mw >>  8) & 0xff), s_trans);
        crf_step(st, a, lane, ec, tg.z, (int)((mw >> 16) & 0xff), s_trans);
        crf_step(st, a, lane, ed, tg.w, (int)((mw >> 24) & 0xff), s_trans);
    }

    // ---- remaining timesteps, 4 per iteration (batched b128 tag / b32 mask)
    for (int t0 = 4; t0 < NT; t0 += 4) {
        float4   ea = ep4[t0], eb = ep4[t0 + 1], ec = ep4[t0 + 2], ed = ep4[t0 + 3];
        int4     tg = *(const int4*)(tp + t0);
        unsigned mw = *(const unsigned*)(mp + t0);

        crf_step(st, a, lane, ea, tg.x, (int)( mw        & 0xff), s_trans);
        crf_step(st, a, lane, eb, tg.y, (int)((mw >>  8) & 0xff), s_trans);
        crf_step(st, a, lane, ec, tg.z, (int)((mw >> 16) & 0xff), s_trans);
        crf_step(st, a, lane, ed, tg.w, (int)((mw >> 24) & 0xff), s_trans);
    }

    // beta kept normalized at every masked step => logZ = logz2 * ln(2).
    if (lane < 16) per_seq[seq] = st.logz2 * LN2 - st.gold;
}

// Deterministic fixed-order reduction: mean over NB per-sequence values.
__global__ __launch_bounds__(256) void crf_reduce_kernel(
    const float* __restrict__ per_seq, float* __restrict__ out)
{
    __shared__ float sm[256];
    float acc = 0.0f;
    for (int i = threadIdx.x; i < NB; i += 256) acc += per_seq[i];
    sm[threadIdx.x] = acc;
    __syncthreads();
    for (int s = 128; s > 0; s >>= 1) {
        if (threadIdx.x < s) sm[threadIdx.x] += sm[threadIdx.x + s];
        __syncthreads();
    }
    if (threadIdx.x == 0) out[0] = sm[0] / (float)NB;
}

extern "C" void kernel_launch(void* const* d_in, const int* in_sizes, int n_in,
                              void* d_out, int out_size, void* d_ws, size_t ws_size,
                              hipStream_t stream) {
    (void)in_sizes; (void)n_in; (void)out_size; (void)ws_size;
    const float*         em    = (const float*)d_in[0];
    const int*           tags  = (const int*)d_in[1];
    const unsigned char* mask  = (const unsigned char*)d_in[2];
    const float*         trans = (const float*)d_in[3];
    float*               out   = (float*)d_out;
    float*               ws    = (float*)d_ws;   // NB floats

    crf_fwd_kernel<<<NB / 128, 256, 0, stream>>>(em, tags, mask, trans, ws);
    crf_reduce_kernel<<<1, 256, 0, stream>>>(ws, out);
}